// PancreaticGCN_35141422415973
// MI455X (gfx1250) — compile-verified
//
#include <hip/hip_runtime.h>

typedef __attribute__((ext_vector_type(2))) float v2f;
typedef __attribute__((ext_vector_type(8))) float v8f;

// ---------------- init / degree kernels ----------------

__global__ void gcn_init_deg(float* __restrict__ deg, int n) {
  int i = blockIdx.x * blockDim.x + threadIdx.x;
  if (i < n) deg[i] = 1.0f;  // self-loop contributes 1 to degree
}

__global__ void gcn_zero(float* __restrict__ p, long long n) {
  long long i = (long long)blockIdx.x * blockDim.x + threadIdx.x;
  if (i < n) p[i] = 0.0f;
}

__global__ void gcn_init_out(float* __restrict__ out, const float* __restrict__ b2,
                             int n3) {
  int i = blockIdx.x * blockDim.x + threadIdx.x;
  if (i < n3) out[i] = b2[i % 3];  // fold bias into accumulator init
}

__global__ void gcn_deg_accum(const int* __restrict__ dst, float* __restrict__ deg,
                              int E) {
  int e = blockIdx.x * blockDim.x + threadIdx.x;
  if (e < E) atomicAdd(&deg[dst[e]], 1.0f);
}

__global__ void gcn_rsqrt(float* __restrict__ deg, int n) {
  int i = blockIdx.x * blockDim.x + threadIdx.x;
  if (i < n) deg[i] = rsqrtf(deg[i]);  // deg >= 1 always (self-loops)
}

// ---------------- GEMM1: h1 = x @ W1 via V_WMMA_F32_16X16X4_F32 ----------------
// One wave computes a 16x128 output tile (8 N-tiles of 16, v8f accumulators).
// A layout (32-bit A 16x4): lanes 0-15 -> M=lane, K=k..k+1 ; lanes 16-31 -> M=lane-16, K=k+2..k+3.
// B layout mirrors with N across lanes. C/D: VGPR v -> row m0 + v + 8*(lane>=16), col = n0 + lane%16.
__global__ __launch_bounds__(256) void gcn_gemm1_wmma(
    const float* __restrict__ x, const float* __restrict__ W,
    float* __restrict__ h, int M) {
  const int lane = threadIdx.x & 31;
  const int wave = threadIdx.x >> 5;
  const int m0 = (blockIdx.x * 4 + wave) * 16;
  if (m0 >= M) return;  // wave-uniform exit; EXEC stays all-ones for WMMA

  const int laneM = lane & 15;
  const int kOff  = (lane >> 4) << 1;  // 0 for lanes 0-15, 2 for lanes 16-31
  int row = m0 + laneM;
  if (row >= M) row = M - 1;  // clamp loads; stores guarded below
  const float* __restrict__ arow = x + (long long)row * 512;

  v8f c[8] = {};

  for (int k = 0; k < 512; k += 4) {
    v2f a;
    a.x = arow[k + kOff];
    a.y = arow[k + kOff + 1];
    const float* __restrict__ w0 = W + (long long)(k + kOff) * 128 + laneM;
    const float* __restrict__ w1 = w0 + 128;
#pragma unroll
    for (int j = 0; j < 8; ++j) {
      v2f b;
      b.x = w0[j * 16];
      b.y = w1[j * 16];
      c[j] = __builtin_amdgcn_wmma_f32_16x16x4_f32(
          /*neg_a=*/false, a, /*neg_b=*/false, b,
          /*c_mod=*/(short)0, c[j], /*reuse_a=*/false, /*reuse_b=*/false);
    }
  }

  const int rbase = m0 + ((lane >> 4) << 3);
#pragma unroll
  for (int j = 0; j < 8; ++j) {
    const int col = j * 16 + laneM;
#pragma unroll
    for (int v = 0; v < 8; ++v) {
      const int r = rbase + v;
      if (r < M) h[(long long)r * 128 + col] = c[j][v];
    }
  }
}

// ---------------- edge aggregation layer 1: wave per edge, float4 per lane ----
__global__ __launch_bounds__(256) void gcn_agg1(
    const int* __restrict__ src, const int* __restrict__ dst,
    const float* __restrict__ dinv, const float* __restrict__ h1,
    float* __restrict__ agg, int E) {
  long long gid = (long long)blockIdx.x * blockDim.x + threadIdx.x;
  int e = (int)(gid >> 5);
  int lane = (int)(gid & 31);
  if (e >= E) return;
  int s = src[e], d = dst[e];
  float nrm = dinv[s] * dinv[d];
  const float4 v = ((const float4*)(h1 + (long long)s * 128))[lane];
  float* op = agg + (long long)d * 128 + lane * 4;
  atomicAdd(op + 0, nrm * v.x);
  atomicAdd(op + 1, nrm * v.y);
  atomicAdd(op + 2, nrm * v.z);
  atomicAdd(op + 3, nrm * v.w);
}

// ---------------- self-loop + bias + relu (non-atomic post-pass) --------------
__global__ void gcn_post1(const float* __restrict__ h1,
                          const float* __restrict__ dinv,
                          const float* __restrict__ b1,
                          float* __restrict__ agg, int total) {
  int t = blockIdx.x * blockDim.x + threadIdx.x;
  if (t >= total) return;
  int i = t >> 7, f = t & 127;
  float di = dinv[i];
  float v = agg[t] + di * di * h1[t] + b1[f];
  agg[t] = v > 0.0f ? v : 0.0f;
}

// ---------------- GEMM2: h2 = y1 @ W2 (N_out = 3, plain VALU) -----------------
__global__ void gcn_gemm2(const float* __restrict__ y1,
                          const float* __restrict__ W2,
                          float* __restrict__ h2, int N) {
  int i = blockIdx.x * blockDim.x + threadIdx.x;
  if (i >= N) return;
  const float* __restrict__ row = y1 + (long long)i * 128;
  float a0 = 0.f, a1 = 0.f, a2 = 0.f;
#pragma unroll 4
  for (int k = 0; k < 128; ++k) {
    float xv = row[k];
    a0 += xv * W2[k * 3 + 0];
    a1 += xv * W2[k * 3 + 1];
    a2 += xv * W2[k * 3 + 2];
  }
  h2[(long long)i * 3 + 0] = a0;
  h2[(long long)i * 3 + 1] = a1;
  h2[(long long)i * 3 + 2] = a2;
}

// ---------------- edge + self-loop aggregation layer 2 ------------------------
__global__ void gcn_agg2(const int* __restrict__ src, const int* __restrict__ dst,
                         const float* __restrict__ dinv,
                         const float* __restrict__ h2,
                         float* __restrict__ out, int E, int N) {
  int t = blockIdx.x * blockDim.x + threadIdx.x;
  if (t < E) {
    int s = src[t], d = dst[t];
    float nrm = dinv[s] * dinv[d];
    atomicAdd(&out[(long long)d * 3 + 0], nrm * h2[(long long)s * 3 + 0]);
    atomicAdd(&out[(long long)d * 3 + 1], nrm * h2[(long long)s * 3 + 1]);
    atomicAdd(&out[(long long)d * 3 + 2], nrm * h2[(long long)s * 3 + 2]);
  } else if (t < E + N) {
    int i = t - E;
    float nn = dinv[i] * dinv[i];
    atomicAdd(&out[(long long)i * 3 + 0], nn * h2[(long long)i * 3 + 0]);
    atomicAdd(&out[(long long)i * 3 + 1], nn * h2[(long long)i * 3 + 1]);
    atomicAdd(&out[(long long)i * 3 + 2], nn * h2[(long long)i * 3 + 2]);
  }
}

extern "C" void kernel_launch(void* const* d_in, const int* in_sizes, int n_in,
                              void* d_out, int out_size, void* d_ws, size_t ws_size,
                              hipStream_t stream) {
  const float* x  = (const float*)d_in[0];   // [N, 512]
  const int*   ei = (const int*)d_in[1];     // [2, E]
  const float* W1 = (const float*)d_in[2];   // [512, 128]
  const float* b1 = (const float*)d_in[3];   // [128]
  const float* W2 = (const float*)d_in[4];   // [128, 3]
  const float* b2 = (const float*)d_in[5];   // [3]
  float* out = (float*)d_out;                // [N, 3]

  const int N = in_sizes[0] / 512;
  const int E = in_sizes[1] / 2;
  const int* src = ei;
  const int* dst = ei + E;

  float* ws   = (float*)d_ws;
  float* deg  = ws;                                  // N floats -> becomes dinv
  float* h1   = ws + (((long long)N + 255) & ~255LL); // N*128
  float* agg1 = h1 + (long long)N * 128;             // N*128 (becomes y1)
  float* h2   = agg1 + (long long)N * 128;           // N*3

  const int B = 256;
  const long long tot1 = (long long)N * 128;
  const long long tE32 = (long long)E * 32;

  gcn_init_deg<<<(N + B - 1) / B, B, 0, stream>>>(deg, N);
  gcn_zero<<<(int)((tot1 + B - 1) / B), B, 0, stream>>>(agg1, tot1);
  gcn_init_out<<<(N * 3 + B - 1) / B, B, 0, stream>>>(out, b2, N * 3);
  gcn_deg_accum<<<(E + B - 1) / B, B, 0, stream>>>(dst, deg, E);
  gcn_rsqrt<<<(N + B - 1) / B, B, 0, stream>>>(deg, N);

  gcn_gemm1_wmma<<<(N + 63) / 64, B, 0, stream>>>(x, W1, h1, N);

  gcn_agg1<<<(int)((tE32 + B - 1) / B), B, 0, stream>>>(src, dst, deg, h1, agg1, E);
  gcn_post1<<<(int)((tot1 + B - 1) / B), B, 0, stream>>>(h1, deg, b1, agg1, (int)tot1);

  gcn_gemm2<<<(N + B - 1) / B, B, 0, stream>>>(agg1, W2, h2, N);
  gcn_agg2<<<(E + N + B - 1) / B, B, 0, stream>>>(src, dst, deg, h2, out, E, N);
}